// ReprojectionLayer_9646496547535
// MI455X (gfx1250) — compile-verified
//
#include <hip/hip_runtime.h>

typedef float v2f __attribute__((ext_vector_type(2)));
typedef float v8f __attribute__((ext_vector_type(8)));

#define G    64
#define G3   (G * G * G)       // 262144
#define NB   2
#define NC   8
#define NJ   23
#define HMH  512
#define HMW  640
#define HW   (HMH * HMW)       // 327680

// One thread per (batch, voxel). Wave32: each wave owns 32 consecutive voxels
// (z fastest -> coalesced output stores). Projection of all 32 voxels x 8
// cameras is done with 4 x V_WMMA_F32_16X16X4_F32 per wave:
//   A (16x4)  = 16 voxel homogeneous coords [x y z 1]
//   B (4x16)  = 4 camera 4x3 matrices packed into columns N = cam*3 + coord
//   D (16x16) = projections, routed to voxel-owner lanes via a 2KB/wave LDS tile
__global__ __launch_bounds__(256) void reproj_wmma_gather_kernel(
    const float* __restrict__ heatmaps,     // [B, C, J, 512, 640]
    const float* __restrict__ center,       // [B, 3]
    const float* __restrict__ cams,         // [B, C, 4, 3]
    float* __restrict__ out)                // [B, J, 64, 64, 64]
{
    __shared__ float projLds[8][32 * 16];   // 8 waves x (32 voxels x 16 cols)

    const int tid   = threadIdx.x;
    const int lane  = tid & 31;
    const int wave  = tid >> 5;
    const int t     = blockIdx.x * 256 + tid;   // global (b, voxel) id
    const int b     = t >> 18;                  // G3 = 2^18
    const int vox   = t & (G3 - 1);

    const int waveBaseT = blockIdx.x * 256 + (tid & ~31);
    const int m16   = lane & 15;                // A-row / B-column this lane feeds
    const int khalf = lane >> 4;                // 0: K={0,1}, 1: K={2,3}

    const float cx = center[b * 3 + 0];
    const float cy = center[b * 3 + 1];
    const float cz = center[b * 3 + 2];

    float acc[NJ];
#pragma unroll
    for (int j = 0; j < NJ; ++j) acc[j] = 0.0f;

    for (int grp = 0; grp < 2; ++grp) {         // cameras grp*4 .. grp*4+3
        // ---- B operand: B[K][N], N = camInGrp*3 + coord (cols 12..15 zero).
        // 32-bit 4x16 layout: VGPR0 holds rows K=0 (lanes 0-15), K=2 (lanes 16-31);
        // VGPR1 holds rows K=1 / K=3.
        v2f bmat;
        {
            float b0 = 0.0f, b1 = 0.0f;
            if (m16 < 12) {
                const int camInGrp = m16 / 3;
                const int coord    = m16 - camInGrp * 3;
                const int cam      = grp * 4 + camInGrp;
                const float* cp = cams + (size_t)((b * NC + cam) * 4) * 3 + coord;
                b0 = cp[(2 * khalf + 0) * 3];   // row K = 2*khalf
                b1 = cp[(2 * khalf + 1) * 3];   // row K = 2*khalf + 1
            }
            bmat.x = b0;
            bmat.y = b1;
        }

        // ---- two WMMAs: h=0 -> wave voxels 0..15, h=1 -> 16..31
#pragma unroll
        for (int h = 0; h < 2; ++h) {
            const int tv = waveBaseT + h * 16 + m16;    // voxel this A-row describes
            const int vv = tv & (G3 - 1);
            const float px = (float)((vv >> 12)       - 32) * 2.0f + cx;
            const float py = (float)(((vv >> 6) & 63) - 32) * 2.0f + cy;
            const float pz = (float)((vv & 63)        - 32) * 2.0f + cz;

            // 32-bit 16x4 A layout: VGPR0 = K0 (lanes 0-15) / K2 (lanes 16-31),
            // VGPR1 = K1 / K3.  Homo vector is [x y z 1].
            v2f amat;
            amat.x = khalf ? pz   : px;
            amat.y = khalf ? 1.0f : py;

            v8f d = {};
            d = __builtin_amdgcn_wmma_f32_16x16x4_f32(
                    /*neg_a=*/false, amat, /*neg_b=*/false, bmat,
                    /*c_mod=*/(short)0, d, /*reuse_a=*/false, /*reuse_b=*/false);

            // D layout: VGPR r, lanes 0-15 -> M=r, lanes 16-31 -> M=r+8; N = lane&15
#pragma unroll
            for (int r = 0; r < 8; ++r) {
                const int M = r + 8 * khalf;
                projLds[wave][(h * 16 + M) * 16 + m16] = d[r];
            }
        }
        __syncthreads();

        // ---- consume projections for this thread's own voxel (= wave lane)
        const float* pl = &projLds[wave][lane * 16];
#pragma unroll
        for (int ci = 0; ci < 4; ++ci) {
            const float pu = pl[ci * 3 + 0];
            const float pv = pl[ci * 3 + 1];
            const float pw = pl[ci * 3 + 2];
            // Fast perspective divide: single v_rcp_f32 + 2 muls. The result
            // only feeds a clamp + half-pixel integer truncation, so ~1 ulp
            // reciprocal is indistinguishable from IEEE division here and
            // avoids the ~10-op div_scale/div_fmas chain per divide.
            const float rw = __builtin_amdgcn_rcpf(pw);
            float u = fminf(fmaxf(pu * rw, 0.0f), (float)(2 * HMW - 1));
            float v = fminf(fmaxf(pv * rw, 0.0f), (float)(2 * HMH - 1));
            int idx = (int)(v * 0.5f) * HMW + (int)(u * 0.5f);
            idx = min(max(idx, 0), HW - 1);     // safety clamp (NaN-proof)

            const int cam = grp * 4 + ci;
            const float* hp = heatmaps + (size_t)((b * NC + cam) * NJ) * HW + idx;
#pragma unroll
            for (int j = 0; j < NJ; ++j)
                acc[j] += hp[(size_t)j * HW];   // 23 independent gathers -> deep MLP
        }
        __syncthreads();
    }

    // mean over 8 cameras; coalesced stores (32 consecutive voxels per wave)
#pragma unroll
    for (int j = 0; j < NJ; ++j)
        out[((size_t)(b * NJ + j) << 18) + vox] = acc[j] * 0.125f;
}

extern "C" void kernel_launch(void* const* d_in, const int* in_sizes, int n_in,
                              void* d_out, int out_size, void* d_ws, size_t ws_size,
                              hipStream_t stream) {
    const float* heatmaps = (const float*)d_in[0];   // [2, 8, 23, 512, 640] f32
    const float* center   = (const float*)d_in[1];   // [2, 3] f32
    const float* cams     = (const float*)d_in[2];   // [2, 8, 4, 3] f32
    float* out            = (float*)d_out;           // [2, 23, 64, 64, 64] f32

    const int totalThreads = NB * G3;                // 524288
    const int block = 256;
    const int grid  = totalThreads / block;          // 2048, exact

    reproj_wmma_gather_kernel<<<grid, block, 0, stream>>>(heatmaps, center, cams, out);
}